// TGRU_88467736363313
// MI455X (gfx1250) — compile-verified
//
#include <hip/hip_runtime.h>
#include <math.h>

// Problem constants (reference: LAB=64, DEMO=32, HID=32, FEAT=8, HEADS=4, HD=2, B=64, T=128)
#define kB    64
#define kT    128
#define kL    64
#define kF    8
#define kHid  32
#define kDemo 32

typedef float v2f __attribute__((ext_vector_type(2)));
typedef float v8f __attribute__((ext_vector_type(8)));

// D = A(16x4 f32) * B(4x16 f32) + C, exact fp32 via CDNA5 matrix pipe.
__device__ __forceinline__ v8f wmma4(v2f a, v2f b, v8f c) {
  return __builtin_amdgcn_wmma_f32_16x16x4_f32(
      /*neg_a=*/false, a, /*neg_b=*/false, b,
      /*c_mod=*/(short)0, c, /*reuse_a=*/false, /*reuse_b=*/false);
}

__device__ __forceinline__ float sigm(float x) { return 1.f / (1.f + __expf(-x)); }

// ---------------------------------------------------------------- demo_h = demo @ demo_W + b
__global__ void k_demo(const float* __restrict__ demo, const float* __restrict__ W,
                       const float* __restrict__ bias, float* __restrict__ dh) {
  int i = blockIdx.x * blockDim.x + threadIdx.x;
  if (i >= kB * kHid) return;
  int b = i >> 5, n = i & 31;
  float acc = bias[n];
#pragma unroll
  for (int k = 0; k < kDemo; ++k) acc += demo[b * kDemo + k] * W[k * kHid + n];
  dh[i] = acc;
}

// ---------------------------------------------------------------- x = lab @ labp_W + b  (WMMA f32)
// M = B*T = 8192, N = 64, K = 64. One wave per 16x16 tile; 4 waves/block cover N.
__global__ void k_labproj(const float* __restrict__ lab, const float* __restrict__ W,
                          const float* __restrict__ bias, float* __restrict__ x) {
  const int wave = threadIdx.x >> 5;
  const int lane = threadIdx.x & 31;
  const int half = lane >> 4, l16 = lane & 15;
  const int m0 = blockIdx.x * 16;
  const int n0 = wave * 16;
  const float* arow = lab + (size_t)(m0 + l16) * kL;
  v8f c = {0, 0, 0, 0, 0, 0, 0, 0};
  for (int kk = 0; kk < kL; kk += 4) {
    int ka = kk + half * 2;                         // lanes 0-15: K=kk,kk+1 ; 16-31: kk+2,kk+3
    v2f a  = {arow[ka], arow[ka + 1]};
    v2f bv = {W[ka * kL + n0 + l16], W[(ka + 1) * kL + n0 + l16]};
    c = wmma4(a, bv, c);
  }
#pragma unroll
  for (int r = 0; r < 8; ++r) {
    int m = m0 + r + half * 8;
    int n = n0 + l16;
    x[(size_t)m * kL + n] = c[r] + bias[n];
  }
}

// ---------------------------------------------------------------- per-channel GRU scan
// grid = L channels, block = B threads (one GRU cell per thread). Weights for the
// channel staged in LDS. Sequential in T by construction; packed b128 feat stores.
__global__ void k_gru(const float* __restrict__ x,
                      const float* __restrict__ Wih, const float* __restrict__ bih,
                      const float* __restrict__ Whh, const float* __restrict__ bhh,
                      float* __restrict__ feat) {
  __shared__ float sWih[24], sBih[24], sBhh[24], sWhh[192];
  const int l = blockIdx.x;
  const int b = threadIdx.x;
  for (int i = threadIdx.x; i < 24; i += kB) {
    sWih[i] = Wih[l * 24 + i];
    sBih[i] = bih[l * 24 + i];
    sBhh[i] = bhh[l * 24 + i];
  }
  for (int i = threadIdx.x; i < 192; i += kB) sWhh[i] = Whh[l * 192 + i];
  __syncthreads();

  float h[kF];
#pragma unroll
  for (int f = 0; f < kF; ++f) h[f] = 0.f;

  for (int t = 0; t < kT; ++t) {
    float xv = x[((size_t)b * kT + t) * kL + l];
    float gh[24];
#pragma unroll
    for (int g = 0; g < 24; ++g) {
      float s = sBhh[g];
#pragma unroll
      for (int f = 0; f < kF; ++f) s += h[f] * sWhh[g * 8 + f];
      gh[g] = s;
    }
#pragma unroll
    for (int f = 0; f < kF; ++f) {
      float r  = sigm(xv * sWih[f]      + sBih[f]      + gh[f]);
      float z  = sigm(xv * sWih[8 + f]  + sBih[8 + f]  + gh[8 + f]);
      float nn = tanhf(xv * sWih[16 + f] + sBih[16 + f] + r * gh[16 + f]);
      h[f] = (1.f - z) * nn + z * h[f];
    }
    float4* fp4 = (float4*)(feat + (((size_t)b * kT + t) * kL + l) * kF);
    fp4[0] = make_float4(h[0], h[1], h[2], h[3]);
    fp4[1] = make_float4(h[4], h[5], h[6], h[7]);
  }
}

// ---------------------------------------------------------------- attention weights
// One block per (l,b). The 4 KB feat tile is staged into LDS with async-to-LDS b128
// transfers (one per lane, no VGPR round trip, tracked by ASYNCcnt). Q,K [T,F] are
// then built in LDS; per head, each wave owns one 16-row s-tile, produces the full
// 128-wide row strip with 8 WMMAs (HD=2 zero-padded to K=4), softmaxes in the C
// layout via half-wave shfl_xor reductions, accumulates the head mean, and streams
// the result out with non-temporal stores (268 MB, write-once, > L2 capacity).
__global__ void k_attn(const float* __restrict__ feat,
                       const float* __restrict__ Wq, const float* __restrict__ bq,
                       const float* __restrict__ Wk, const float* __restrict__ bk,
                       float* __restrict__ attn) {
  __shared__ float sF[kT * kF];                   // feat tile [128][8]
  __shared__ float sQ[kT][kF], sK[kT][kF];
  __shared__ float sWq[64], sWk[64], sbq[8], sbk[8];
  const int l = blockIdx.x, b = blockIdx.y;
  const int tid = threadIdx.x;

  // Async stage of the feat tile: thread i moves 16B of row s = i>>1, half j = i&1.
  {
    int s = tid >> 1, j = tid & 1;
    const float* gsrc = feat + (((size_t)b * kT + s) * kL + l) * kF + j * 4;
    unsigned ldsOff = (unsigned)(uintptr_t)(&sF[s * kF + j * 4]);  // addrspace(3) byte offset
    asm volatile("global_load_async_to_lds_b128 %0, %1, off"
                 :: "v"(ldsOff), "v"(gsrc) : "memory");
  }
  if (tid < 64) { sWq[tid] = Wq[l * 64 + tid]; sWk[tid] = Wk[l * 64 + tid]; }
  if (tid < 8)  { sbq[tid] = bq[l * 8 + tid];  sbk[tid] = bk[l * 8 + tid]; }
  asm volatile("s_wait_asynccnt 0x0" ::: "memory");
  __syncthreads();

  for (int i = tid; i < kT * kF; i += 256) {
    int s = i >> 3, e = i & 7;
    const float* fp = &sF[s * kF];
    float q = sbq[e], k = sbk[e];
#pragma unroll
    for (int f = 0; f < kF; ++f) { q += fp[f] * sWq[e * 8 + f]; k += fp[f] * sWk[e * 8 + f]; }
    sQ[s][e] = q;
    sK[s][e] = k;
  }
  __syncthreads();

  const int wv = tid >> 5, lane = tid & 31, half = lane >> 4, l16 = lane & 15;
  const int s0 = wv * 16;                         // 8 waves cover the 8 s-tiles
  const float scale = 0.70710678118654752f;       // 1/sqrt(HD)

  float macc[8][8];
#pragma unroll
  for (int r = 0; r < 8; ++r)
#pragma unroll
    for (int t = 0; t < 8; ++t) macc[r][t] = 0.f;

  for (int hh = 0; hh < 4; ++hh) {
    v2f a = {0.f, 0.f};                           // lanes 16-31 feed K=2,3 -> zero pad
    if (half == 0) {
      a.x = sQ[s0 + l16][2 * hh] * scale;
      a.y = sQ[s0 + l16][2 * hh + 1] * scale;
    }
    v8f C[8];
#pragma unroll
    for (int tt = 0; tt < 8; ++tt) {
      v2f bv = {0.f, 0.f};
      if (half == 0) {
        bv.x = sK[tt * 16 + l16][2 * hh];
        bv.y = sK[tt * 16 + l16][2 * hh + 1];
      }
      v8f cz = {0, 0, 0, 0, 0, 0, 0, 0};
      C[tt] = wmma4(a, bv, cz);
    }
    // softmax over t per row; row (s0+r+half*8) lives in the 16 lanes of this half.
#pragma unroll
    for (int r = 0; r < 8; ++r) {
      float m = -3.4e38f;
#pragma unroll
      for (int tt = 0; tt < 8; ++tt) m = fmaxf(m, C[tt][r]);
#pragma unroll
      for (int mk = 8; mk >= 1; mk >>= 1) m = fmaxf(m, __shfl_xor(m, mk, 32));
      float p[8], sum = 0.f;
#pragma unroll
      for (int tt = 0; tt < 8; ++tt) { p[tt] = __expf(C[tt][r] - m); sum += p[tt]; }
#pragma unroll
      for (int mk = 8; mk >= 1; mk >>= 1) sum += __shfl_xor(sum, mk, 32);
      float inv = 0.25f / sum;                    // head mean folded in
#pragma unroll
      for (int tt = 0; tt < 8; ++tt) macc[r][tt] += p[tt] * inv;
    }
  }

  // attn[b][s][l][t] — non-temporal (write-once, > L2 capacity)
#pragma unroll
  for (int r = 0; r < 8; ++r) {
    int s = s0 + r + half * 8;
    size_t base = (((size_t)b * kT + s) * kL + l) * kT;
#pragma unroll
    for (int tt = 0; tt < 8; ++tt)
      __builtin_nontemporal_store(macc[r][tt], attn + base + tt * 16 + l16);
  }
}

// ---------------------------------------------------------------- out = [demo_h | flat] @ out_W + b
// M = 8192, N = 32, K = 544. One wave per 16x16 tile (1024 tiles), K-loop of 136 WMMAs.
__global__ void k_out(const float* __restrict__ dh, const float* __restrict__ feat,
                      const float* __restrict__ W, const float* __restrict__ bias,
                      float* __restrict__ out) {
  const int wv = threadIdx.x >> 5, lane = threadIdx.x & 31;
  const int half = lane >> 4, l16 = lane & 15;
  const int w = blockIdx.x * 8 + wv;              // 0..1023
  const int m0 = (w >> 1) * 16, n0 = (w & 1) * 16;
  const int m = m0 + l16;
  const float* drow = dh + (size_t)(m >> 7) * kHid;     // m / T = batch index
  const float* frow = feat + (size_t)m * (kL * kF);     // flat feature row (512)
  v8f c = {0, 0, 0, 0, 0, 0, 0, 0};
  for (int kk = 0; kk < kDemo + kL * kF; kk += 4) {
    int ka = kk + half * 2;
    v2f a;
    if (ka < kDemo) { a.x = drow[ka];         a.y = drow[ka + 1]; }
    else            { a.x = frow[ka - kDemo]; a.y = frow[ka - kDemo + 1]; }
    v2f bv = {W[ka * kHid + n0 + l16], W[(ka + 1) * kHid + n0 + l16]};
    c = wmma4(a, bv, c);
  }
#pragma unroll
  for (int r = 0; r < 8; ++r) {
    int mm = m0 + r + half * 8;
    int n = n0 + l16;
    out[(size_t)mm * kHid + n] = c[r] + bias[n];
  }
}

// ----------------------------------------------------------------
extern "C" void kernel_launch(void* const* d_in, const int* in_sizes, int n_in,
                              void* d_out, int out_size, void* d_ws, size_t ws_size,
                              hipStream_t stream) {
  const float* lab   = (const float*)d_in[0];
  const float* demo  = (const float*)d_in[1];
  const float* demoW = (const float*)d_in[2];
  const float* demob = (const float*)d_in[3];
  const float* labpW = (const float*)d_in[4];
  const float* labpb = (const float*)d_in[5];
  const float* Wih   = (const float*)d_in[6];
  const float* bih   = (const float*)d_in[7];
  const float* Whh   = (const float*)d_in[8];
  const float* bhh   = (const float*)d_in[9];
  const float* Wq    = (const float*)d_in[10];
  const float* bq    = (const float*)d_in[11];
  const float* Wk    = (const float*)d_in[12];
  const float* bk    = (const float*)d_in[13];
  const float* outW  = (const float*)d_in[14];
  const float* outb  = (const float*)d_in[15];

  float* out  = (float*)d_out;                          // [B,T,HID]
  float* attn = out + (size_t)kB * kT * kHid;           // [B,T,L,T]

  // Workspace layout (floats): demo_h[2048] | x[524288] | feat[4194304] (~18.9 MB)
  float* ws   = (float*)d_ws;
  float* dh   = ws;
  float* x    = dh + kB * kHid;
  float* feat = x + (size_t)kB * kT * kL;

  k_demo   <<<dim3((kB * kHid + 255) / 256), dim3(256), 0, stream>>>(demo, demoW, demob, dh);
  k_labproj<<<dim3(kB * kT / 16),            dim3(128), 0, stream>>>(lab, labpW, labpb, x);
  k_gru    <<<dim3(kL),                      dim3(kB),  0, stream>>>(x, Wih, bih, Whh, bhh, feat);
  k_attn   <<<dim3(kL, kB),                  dim3(256), 0, stream>>>(feat, Wq, bq, Wk, bk, attn);
  k_out    <<<dim3(kB * kT * 2 / 16 / 8),    dim3(256), 0, stream>>>(dh, feat, outW, outb, out);
}